// BubblePredictor_52132313038938
// MI455X (gfx1250) — compile-verified
//
#include <hip/hip_runtime.h>

// GRU: B=128, T=128, I=12, H=512, fp32 (recurrence precision -> f32 WMMA).
// 1 init + 128 fused step kernels + 1 final kernel, all on `stream`.
//
// Step kernel t (grid = 32 WGs x 256 thr): WG owns a 16-wide column tile of H;
// wave w owns batch rows [16w,16w+16). Per wave: 4 f32 16x16 accum tiles
// (r, z, n_x, n_h) built from V_WMMA_F32_16X16X4_F32. W_hh tiles are streamed
// through LDS with GLOBAL_LOAD_ASYNC_TO_LDS_B128 in a 2-deep double-buffered
// pipeline synchronized with s_wait_asynccnt + workgroup barriers.

typedef float v8f __attribute__((ext_vector_type(8)));
typedef float v2f __attribute__((ext_vector_type(2)));
typedef int   v4i __attribute__((ext_vector_type(4)));

typedef __attribute__((address_space(1))) v4i*  gv4i_t;   // global int4*
typedef __attribute__((address_space(3))) v4i*  lv4i_t;   // LDS int4*
typedef __attribute__((address_space(1))) void* gvoid_t;
typedef __attribute__((address_space(3))) void* lvoid_t;

#define BB 128
#define TT 128
#define II 12
#define HH 512

#define KC   128                // K-chunk per pipeline stage
#define LROW (KC + 4)           // padded LDS row stride (dwords) -> no bank conflicts
#define LBUF (48 * LROW)        // one buffer: 3 gates x 16 rows

#if defined(__has_builtin)
#  if __has_builtin(__builtin_amdgcn_global_load_async_to_lds_b128)
#    define HAS_ASYNC_LDS 1
#  endif
#endif
#ifndef HAS_ASYNC_LDS
#  define HAS_ASYNC_LDS 0
#endif

#define WMMA_F32(a, b, c) \
    __builtin_amdgcn_wmma_f32_16x16x4_f32(false, (a), false, (b), (short)0, (c), false, false)

__device__ __forceinline__ float sigmoid_f(float x) {
    return 1.0f / (1.0f + __expf(-x));
}
__device__ __forceinline__ float tanh_f(float x) {
    float e = __expf(-2.0f * x);
    return (1.0f - e) / (1.0f + e);
}

// Wait until per-wave ASYNCcnt <= N (async loads retire in order).
__device__ __forceinline__ void wait_async_le6() {
#if HAS_ASYNC_LDS
    asm volatile("s_wait_asynccnt 0x6" ::: "memory");
#endif
}
__device__ __forceinline__ void wait_async_le0() {
#if HAS_ASYNC_LDS
    asm volatile("s_wait_asynccnt 0x0" ::: "memory");
#endif
}

// Stage one K-chunk of the 3 gate tiles of W_hh into LDS.
// tid -> row = tid>>4 (0..15), col4 = (tid&15)*4 ; 6 x b128 per thread.
__device__ __forceinline__ void stage_whh(const float* __restrict__ Whh,
                                          float* __restrict__ dst,
                                          int j0, int kc, int tid) {
    const int row = tid >> 4;
    const int c4  = (tid & 15) * 4;
#pragma unroll
    for (int g = 0; g < 3; ++g) {
#pragma unroll
        for (int hf = 0; hf < 2; ++hf) {
            const float* gsrc = Whh + (g * HH + j0 + row) * HH + kc + hf * 64 + c4;
            float*       ldst = dst + (g * 16 + row) * LROW + hf * 64 + c4;
#if HAS_ASYNC_LDS
            __builtin_amdgcn_global_load_async_to_lds_b128(
                (gv4i_t)(gvoid_t)(void*)gsrc,
                (lv4i_t)(lvoid_t)(void*)ldst, 0, 0);
#else
            *(float4*)ldst = *(const float4*)gsrc;
#endif
        }
    }
}

__global__ __launch_bounds__(256) void gru_step_kernel(
    const float* __restrict__ hist,   // [B, T, I]
    const float* __restrict__ Wih,    // [3H, I]
    const float* __restrict__ Whh,    // [3H, H]
    const float* __restrict__ Wpred,  // [2, H]
    const float* __restrict__ h_in,   // [B, H]  = h_t
    float* __restrict__ h_out,        // [B, H]  = h_{t+1}
    float* __restrict__ logits,       // [B, T, 2]
    int t)
{
    __shared__ float wl[2 * LBUF];    // double-buffered W_hh tile

    const int tid  = threadIdx.x;
    const int lane = tid & 31;
    const int wave = tid >> 5;            // 0..7  -> batch (M) tile
    const int j0   = blockIdx.x * 16;     // column (N) tile base
    const int b0   = wave * 16;
    const int nrow = lane & 15;           // A row / B,C column within tile
    const int hi   = lane >> 4;           // K/M half selector (ISA layout)

    // Pipeline prologue: async-stage chunks 0 and 1 (overlaps x-part WMMAs).
    stage_whh(Whh, wl, j0, 0 * KC, tid);
    stage_whh(Whh, wl + LBUF, j0, 1 * KC, tid);

    v8f acc_r  = {};   // r gate (x + h parts)
    v8f acc_z  = {};   // z gate (x + h parts)
    v8f acc_in = {};   // n gate, x-part
    v8f acc_hn = {};   // n gate, h-part (scaled by r in epilogue)

    const int    brow = b0 + nrow;
    const float* xrow = hist + (brow * TT + t) * II + 2 * hi;

    // ---- x-part: K=12 as three 16x16x4 WMMAs per gate ----
#pragma unroll
    for (int k0 = 0; k0 < II; k0 += 4) {
        v2f a  = *(const v2f*)(xrow + k0);
        v2f br = *(const v2f*)(Wih + (0 * HH + j0 + nrow) * II + k0 + 2 * hi);
        v2f bz = *(const v2f*)(Wih + (1 * HH + j0 + nrow) * II + k0 + 2 * hi);
        v2f bn = *(const v2f*)(Wih + (2 * HH + j0 + nrow) * II + k0 + 2 * hi);
        acc_r  = WMMA_F32(a, br, acc_r);
        acc_z  = WMMA_F32(a, bz, acc_z);
        acc_in = WMMA_F32(a, bn, acc_in);
    }

    // ---- h-part: K=512 in 4 chunks, double-buffered async pipeline ----
#pragma unroll 1
    for (int i = 0; i < 4; ++i) {
        if (i < 3) wait_async_le6(); else wait_async_le0();  // chunk i landed
        __syncthreads();                                     // publish to all waves

        const float* bufb = wl + (i & 1) * LBUF;
        const float* pa = h_in + brow * HH + i * KC + 2 * hi;
        const float* p0 = bufb + (0 * 16 + nrow) * LROW + 2 * hi;
        const float* p1 = bufb + (1 * 16 + nrow) * LROW + 2 * hi;
        const float* p2 = bufb + (2 * 16 + nrow) * LROW + 2 * hi;
#pragma unroll
        for (int k0 = 0; k0 < KC; k0 += 4) {
            v2f a  = *(const v2f*)(pa + k0);
            v2f br = *(const v2f*)(p0 + k0);
            v2f bz = *(const v2f*)(p1 + k0);
            v2f bn = *(const v2f*)(p2 + k0);
            acc_r  = WMMA_F32(a, br, acc_r);
            acc_z  = WMMA_F32(a, bz, acc_z);
            acc_hn = WMMA_F32(a, bn, acc_hn);
        }

        if (i < 2) {
            __syncthreads();                 // all waves done reading this buffer
            stage_whh(Whh, wl + (i & 1) * LBUF, j0, (i + 2) * KC, tid);
        }
    }

    // ---- gate epilogue: C layout VGPR i -> M = i + 8*hi, N = lane&15 ----
#pragma unroll
    for (int i = 0; i < 8; ++i) {
        int b = b0 + i + 8 * hi;
        int j = j0 + nrow;
        float r  = sigmoid_f(acc_r[i]);
        float z  = sigmoid_f(acc_z[i]);
        float n  = tanh_f(acc_in[i] + r * acc_hn[i]);
        float hp = h_in[b * HH + j];
        h_out[b * HH + j] = (1.0f - z) * n + z * hp;
    }

    // ---- logit head for the PREVIOUS step's h (h_in fully complete) ----
    if (blockIdx.x == 0 && t > 0) {
        int b = tid >> 1, k = tid & 1;
        const float* hr = h_in + b * HH;
        const float* wp = Wpred + k * HH;
        float s = 0.0f;
        for (int q = 0; q < HH; q += 4) {
            float4 hv = *(const float4*)(hr + q);
            float4 wv = *(const float4*)(wp + q);
            s += hv.x * wv.x + hv.y * wv.y + hv.z * wv.z + hv.w * wv.w;
        }
        logits[b * (TT * 2) + (t - 1) * 2 + k] = s;
    }
}

__global__ __launch_bounds__(256) void gru_init_kernel(float* __restrict__ h0) {
    int stride = gridDim.x * 256;
    for (int idx = blockIdx.x * 256 + threadIdx.x; idx < BB * HH; idx += stride)
        h0[idx] = 0.0f;
}

__global__ __launch_bounds__(256) void gru_final_kernel(
    const float* __restrict__ h_fin, const float* __restrict__ Wpred,
    float* __restrict__ logits, float* __restrict__ h_last)
{
    if (blockIdx.x == 0) {  // logits for t = T-1
        int b = threadIdx.x >> 1, k = threadIdx.x & 1;
        const float* hr = h_fin + b * HH;
        const float* wp = Wpred + k * HH;
        float s = 0.0f;
        for (int q = 0; q < HH; q += 4) {
            float4 hv = *(const float4*)(hr + q);
            float4 wv = *(const float4*)(wp + q);
            s += hv.x * wv.x + hv.y * wv.y + hv.z * wv.z + hv.w * wv.w;
        }
        logits[b * (TT * 2) + (TT - 1) * 2 + k] = s;
    }
    int stride = gridDim.x * 256;
    for (int idx = blockIdx.x * 256 + threadIdx.x; idx < BB * HH; idx += stride)
        h_last[idx] = h_fin[idx];
}

extern "C" void kernel_launch(void* const* d_in, const int* in_sizes, int n_in,
                              void* d_out, int out_size, void* d_ws, size_t ws_size,
                              hipStream_t stream) {
    const float* hist  = (const float*)d_in[0];   // [128,128,12]
    const float* Wih   = (const float*)d_in[1];   // [1536,12]
    const float* Whh   = (const float*)d_in[2];   // [1536,512]
    const float* Wpred = (const float*)d_in[3];   // [2,512]

    float* logits = (float*)d_out;                 // [128,128,2]
    float* h_last = (float*)d_out + BB * TT * 2;   // [128,512]

    float* buf0 = (float*)d_ws;            // h ping  (B*H floats)
    float* buf1 = buf0 + BB * HH;          // h pong

    gru_init_kernel<<<dim3(64), dim3(256), 0, stream>>>(buf0);

    for (int t = 0; t < TT; ++t) {
        float* hin  = (t & 1) ? buf1 : buf0;
        float* hout = (t & 1) ? buf0 : buf1;
        gru_step_kernel<<<dim3(HH / 16), dim3(256), 0, stream>>>(
            hist, Wih, Whh, Wpred, hin, hout, logits, t);
    }
    // after t = 127 (odd), final h lives in buf0
    gru_final_kernel<<<dim3(64), dim3(256), 0, stream>>>(buf0, Wpred, logits, h_last);
}